// LQE_64845416235742
// MI455X (gfx1250) — compile-verified
//
#include <hip/hip_runtime.h>
#include <math.h>

typedef __attribute__((ext_vector_type(2))) float v2f;
typedef __attribute__((ext_vector_type(8))) float v8f;

#define NBINS 33
#define NC    4
#define INDIM 20     // NC * (K+1)
#define HID   64
#define BLOCK 256
#define WAVES (BLOCK / 32)

// Fused: softmax/top4 stats -> [16x20]@[20x64] via v_wmma_f32_16x16x4_f32
// -> ReLU -> @W2 (shfl reduce) -> broadcast-add into scores.
__global__ __launch_bounds__(BLOCK) void lqe_fused_kernel(
    const float* __restrict__ scores,
    const float* __restrict__ pred,
    const float* __restrict__ W1,
    const float* __restrict__ b1,
    const float* __restrict__ W2,
    const float* __restrict__ b2,
    float* __restrict__ out,
    int npos)
{
    __shared__ float sstat[WAVES][32 * INDIM];  // per-wave 32 stat rows
    __shared__ float squal[WAVES][32];          // per-wave quality scores

    const int tid  = threadIdx.x;
    const int wave = tid >> 5;
    const int lane = tid & 31;
    const int n    = lane & 15;   // column / N index within tile
    const int h    = lane >> 4;   // lane half (K split for A/B operands)

    const int pos  = blockIdx.x * BLOCK + tid;
    const int cpos = (pos < npos) ? pos : (npos - 1);  // clamp: keep EXEC full

    // ---------------- per-position stats: stable softmax, top-4, mean ------
    {
        const float4* p4 = (const float4*)(pred + (size_t)cpos * (NC * NBINS));
        float v[NC * NBINS];
#pragma unroll
        for (int i = 0; i < (NC * NBINS) / 4; ++i) {   // 33 aligned float4 loads
            float4 t = p4[i];
            v[4*i+0] = t.x; v[4*i+1] = t.y; v[4*i+2] = t.z; v[4*i+3] = t.w;
        }
        float stat[INDIM];
#pragma unroll
        for (int c = 0; c < NC; ++c) {
            float t0 = -INFINITY, t1 = -INFINITY, t2 = -INFINITY, t3 = -INFINITY;
#pragma unroll
            for (int j = 0; j < NBINS; ++j) {          // branchless top-4 insert
                float x  = v[c*NBINS + j];
                float l0 = fminf(t0, x);  t0 = fmaxf(t0, x);
                float l1 = fminf(t1, l0); t1 = fmaxf(t1, l0);
                float l2 = fminf(t2, l1); t2 = fmaxf(t2, l1);
                t3 = fmaxf(t3, l2);
            }
            float s = 0.f;
#pragma unroll
            for (int j = 0; j < NBINS; ++j)            // Σ exp(x - max)
                s += __expf(v[c*NBINS + j] - t0);
            const float inv = 1.0f / s;
            const float p0 = inv;                       // exp(0)/s
            const float p1 = __expf(t1 - t0) * inv;
            const float p2 = __expf(t2 - t0) * inv;
            const float p3 = __expf(t3 - t0) * inv;
            stat[5*c+0] = p0; stat[5*c+1] = p1;
            stat[5*c+2] = p2; stat[5*c+3] = p3;
            stat[5*c+4] = 0.25f * (p0 + p1 + p2 + p3);
        }
        float* dst = &sstat[wave][lane * INDIM];
#pragma unroll
        for (int i = 0; i < INDIM; ++i) dst[i] = stat[i];
    }
    __syncthreads();

    // ---------------- layer 1: [16x20] @ W1[20x64] on the matrix pipe ------
    // B (4x16 K-chunk, N-tile t): lane half h supplies K rows 2h, 2h+1.
    v2f Bm[5][4];
#pragma unroll
    for (int k = 0; k < 5; ++k) {
        const int row = 4*k + 2*h;
#pragma unroll
        for (int t = 0; t < 4; ++t) {
            const int col = 16*t + n;
            v2f b; b.x = W1[row*HID + col]; b.y = W1[(row+1)*HID + col];
            Bm[k][t] = b;
        }
    }
    float b1v[4], w2v[4];
#pragma unroll
    for (int t = 0; t < 4; ++t) { b1v[t] = b1[16*t + n]; w2v[t] = W2[16*t + n]; }
    const float b2s = b2[0];

#pragma unroll
    for (int g = 0; g < 2; ++g) {                 // two 16-row groups per wave
        v8f acc0 = {0.f,0.f,0.f,0.f,0.f,0.f,0.f,0.f};
        v8f acc1 = acc0, acc2 = acc0, acc3 = acc0;
        const int r = g*16 + n;                   // stat row owned by this lane
#pragma unroll
        for (int k = 0; k < 5; ++k) {
            // A (16x4 f32): lanes 0-15 K=0,1 ; lanes 16-31 K=2,3 (even offset -> b64)
            const float2 af = *(const float2*)&sstat[wave][r*INDIM + 4*k + 2*h];
            v2f a; a.x = af.x; a.y = af.y;
            acc0 = __builtin_amdgcn_wmma_f32_16x16x4_f32(false, a, false, Bm[k][0], (short)0, acc0, false, false);
            acc1 = __builtin_amdgcn_wmma_f32_16x16x4_f32(false, a, false, Bm[k][1], (short)0, acc1, false, false);
            acc2 = __builtin_amdgcn_wmma_f32_16x16x4_f32(false, a, false, Bm[k][2], (short)0, acc2, false, false);
            acc3 = __builtin_amdgcn_wmma_f32_16x16x4_f32(false, a, false, Bm[k][3], (short)0, acc3, false, false);
        }
        // ---------------- bias + ReLU + layer 2 dot + 16-lane reduce -------
#pragma unroll
        for (int r8 = 0; r8 < 8; ++r8) {          // D row = r8 + 8h (+16g)
            float q = fmaxf(acc0[r8] + b1v[0], 0.f) * w2v[0]
                    + fmaxf(acc1[r8] + b1v[1], 0.f) * w2v[1]
                    + fmaxf(acc2[r8] + b1v[2], 0.f) * w2v[2]
                    + fmaxf(acc3[r8] + b1v[3], 0.f) * w2v[3];
            q += __shfl_xor(q, 1, 16);
            q += __shfl_xor(q, 2, 16);
            q += __shfl_xor(q, 4, 16);
            q += __shfl_xor(q, 8, 16);
            if (n == r8) squal[wave][g*16 + 8*h + r8] = q + b2s;
        }
    }
    __syncthreads();

    // ---------------- broadcast-add: out = scores + quality (80/pos) -------
    const int waveBase = blockIdx.x * BLOCK + wave * 32;
    const float4* s4 = (const float4*)scores;
    float4*       o4 = (float4*)out;
    const size_t base4 = (size_t)waveBase * 20;    // 20 float4 per position
#pragma unroll
    for (int i = 0; i < 20; ++i) {
        const int idx = i*32 + lane;               // 0..639, coalesced
        const int p   = idx / 20;                  // position within wave
        if (waveBase + p < npos) {
            const float qq = squal[wave][p];
            float4 sv = s4[base4 + idx];
            float4 ov; ov.x = sv.x + qq; ov.y = sv.y + qq;
                       ov.z = sv.z + qq; ov.w = sv.w + qq;
            o4[base4 + idx] = ov;
        }
    }
}

extern "C" void kernel_launch(void* const* d_in, const int* in_sizes, int n_in,
                              void* d_out, int out_size, void* d_ws, size_t ws_size,
                              hipStream_t stream) {
    (void)n_in; (void)out_size; (void)d_ws; (void)ws_size;
    const float* scores = (const float*)d_in[0];  // (B, L, 80)
    const float* pred   = (const float*)d_in[1];  // (B, L, 4, 33)
    const float* W1     = (const float*)d_in[2];  // (20, 64)
    const float* b1     = (const float*)d_in[3];  // (64,)
    const float* W2     = (const float*)d_in[4];  // (64, 1)
    const float* b2     = (const float*)d_in[5];  // (1,)
    float* out = (float*)d_out;                   // (B, L, 80)

    const int npos   = in_sizes[0] / 80;          // B*L = 320000
    const int blocks = (npos + BLOCK - 1) / BLOCK;
    lqe_fused_kernel<<<blocks, BLOCK, 0, stream>>>(scores, pred, W1, b1, W2, b2,
                                                   out, npos);
}